// SPIKE_layer_43465069036188
// MI455X (gfx1250) — compile-verified
//
#include <hip/hip_runtime.h>

// Spiking integrate-and-fire scan over T=8 timesteps.
// x: [B=16, T=8, C=256, H=56, W=56] f32; out: same shape.
// Pure streaming problem (~822 MB traffic, ~0.4 GFLOP) -> HBM-bound (~35us floor @ 23.3 TB/s).
// Strategy: float4 per lane, double-buffered async global->LDS staging
// (CDNA5 global_load_async_to_lds_b128 + s_wait_asynccnt pipelining),
// compute scan in registers, non-temporal b128 loads/stores (no L2 pollution,
// data is touched exactly once).

typedef __attribute__((ext_vector_type(4))) float v4f;

#define T_STEPS 8
#define S_ELEMS 802816u      // C*H*W = 256*56*56
#define N4_TOTAL 3211264u    // B*S/4 float4 units
#define BLOCKS 3136          // N4_TOTAL == 4 * BLOCKS * THREADS (exact)
#define THREADS 256
#define ITERS 4

__device__ __forceinline__ void async_ld_b128(unsigned lds_byte_addr,
                                              const float* base,
                                              unsigned byte_off) {
  // GVS mode: mem_addr = SGPR_U64(base) + VGPR_I32(byte_off); LDS dst from VGPR.
  // Non-temporal: stream-once input, keep it out of the way in L2.
  asm volatile("global_load_async_to_lds_b128 %0, %1, %2 th:TH_LOAD_NT"
               :
               : "v"(lds_byte_addr), "v"(byte_off), "s"(base)
               : "memory");
}

__global__ __launch_bounds__(THREADS) void spike_scan_kernel(
    const float* __restrict__ x, const float* __restrict__ th_inner,
    const float* __restrict__ th_outer, float* __restrict__ out) {
  __shared__ v4f tile[2][T_STEPS][THREADS];
  const unsigned tid = threadIdx.x;

  // Uniform threshold prep (scalar loads): inner[t]*T, outer[t]*T
  float thin[T_STEPS], amp[T_STEPS];
#pragma unroll
  for (int t = 0; t < T_STEPS; ++t) {
    thin[t] = th_inner[t] * 8.0f;
    amp[t]  = th_outer[t] * 8.0f;
  }

  const unsigned stride = BLOCKS * THREADS;
  unsigned i = blockIdx.x * THREADS + tid;

  const unsigned lds_base = (unsigned)(unsigned long long)&tile[0][0][tid];
  const unsigned lds_buf_stride = T_STEPS * THREADS * sizeof(v4f);

  // Prologue: stage tile for iteration 0 into buffer 0.
  {
    unsigned f = i * 4u;
    unsigned b = f / S_ELEMS;
    unsigned s = f - b * S_ELEMS;
    unsigned e0 = (b * 8u) * S_ELEMS + s;
#pragma unroll
    for (int t = 0; t < T_STEPS; ++t)
      async_ld_b128(lds_base + (unsigned)t * (THREADS * sizeof(v4f)), x,
                    (e0 + (unsigned)t * S_ELEMS) * 4u);
  }

  unsigned buf = 0;
#pragma unroll 1
  for (int it = 0; it < ITERS; ++it, i += stride, buf ^= 1u) {
    // Prefetch next tile into the other buffer, then retire current tile.
    if (it < ITERS - 1) {
      unsigned inx = i + stride;
      unsigned fn = inx * 4u;
      unsigned bn = fn / S_ELEMS;
      unsigned sn = fn - bn * S_ELEMS;
      unsigned en = (bn * 8u) * S_ELEMS + sn;
      unsigned lds1 = lds_base + (buf ^ 1u) * lds_buf_stride;
#pragma unroll
      for (int t = 0; t < T_STEPS; ++t)
        async_ld_b128(lds1 + (unsigned)t * (THREADS * sizeof(v4f)), x,
                      (en + (unsigned)t * S_ELEMS) * 4u);
      // 16 outstanding, in-order completion: <=8 means current tile is done.
      asm volatile("s_wait_asynccnt 0x8" ::: "memory");
    } else {
      asm volatile("s_wait_asynccnt 0x0" ::: "memory");
    }

    unsigned f = i * 4u;
    unsigned b = f / S_ELEMS;
    unsigned s = f - b * S_ELEMS;
    unsigned e0 = (b * 8u) * S_ELEMS + s;

    v4f mem = {0.0f, 0.0f, 0.0f, 0.0f};
#pragma unroll
    for (int t = 0; t < T_STEPS; ++t) {
      v4f xt = tile[buf][t][tid];   // ds_load_b128
      mem += xt;
      v4f sp;
      sp.x = (mem.x >= thin[t]) ? amp[t] : 0.0f;
      sp.y = (mem.y >= thin[t]) ? amp[t] : 0.0f;
      sp.z = (mem.z >= thin[t]) ? amp[t] : 0.0f;
      sp.w = (mem.w >= thin[t]) ? amp[t] : 0.0f;
      mem -= sp;
      // Streaming output: non-temporal b128 store.
      __builtin_nontemporal_store(sp, (v4f*)(out + (e0 + (unsigned)t * S_ELEMS)));
    }
  }
}

extern "C" void kernel_launch(void* const* d_in, const int* in_sizes, int n_in,
                              void* d_out, int out_size, void* d_ws, size_t ws_size,
                              hipStream_t stream) {
  const float* x        = (const float*)d_in[0];
  const float* th_inner = (const float*)d_in[1];
  const float* th_outer = (const float*)d_in[2];
  // d_in[3] is the unused scalar `t`.
  float* out = (float*)d_out;
  (void)in_sizes; (void)n_in; (void)out_size; (void)d_ws; (void)ws_size;

  spike_scan_kernel<<<dim3(BLOCKS), dim3(THREADS), 0, stream>>>(
      x, th_inner, th_outer, out);
}